// resNet18_2705829396575
// MI455X (gfx1250) — compile-verified
//
#include <hip/hip_runtime.h>
#include <hip/hip_bf16.h>

// ---------------------------------------------------------------------------
// Model dims (match reference)
// ---------------------------------------------------------------------------
#define BATCH 16
#define KSZ   3
#define FILT  6
#define NPIX4 (86 * 86)          // pixels per image after 5 LC layers
#define BN_EPS 1e-3f

typedef __attribute__((ext_vector_type(2))) float v2f;
typedef __attribute__((ext_vector_type(8))) float v8f;

// ---------------------------------------------------------------------------
// LocallyConnected2D (+ fused consumption of previous layer's BN+ReLU) and
// per-channel sum / sum-of-squares accumulation for this layer's BN stats.
// Thread t -> (pixel p = t>>4, batch b = t&15): 16 consecutive lanes share a
// pixel so the per-pixel weight loads broadcast (one L2 request per weight).
// ---------------------------------------------------------------------------
__global__ void __launch_bounds__(256) lc_kernel(
    const float* __restrict__ in,      // prev activations (raw pre-BN y, or x)
    const float* __restrict__ ss_in,   // prev layer scale[6], shift[6]
    int use_bn,                        // 0 for raw input x
    const float* __restrict__ w,       // [oh*ow, 9*cin, 6]
    const float* __restrict__ bias,    // [oh*ow, 6]
    float* __restrict__ out,           // [B, oh, ow, 6] raw pre-BN
    float* __restrict__ stats,         // [12]: sum[6], sumsq[6]
    int h, int w_in, int cin, int oh, int ow)
{
    __shared__ float s_sum[FILT];
    __shared__ float s_ssq[FILT];
    if (threadIdx.x < FILT) { s_sum[threadIdx.x] = 0.f; s_ssq[threadIdx.x] = 0.f; }
    __syncthreads();

    const int t    = blockIdx.x * blockDim.x + threadIdx.x;
    const int npix = oh * ow;
    const bool active = t < npix * BATCH;

    if (active) {
        const int b  = t & (BATCH - 1);
        const int p  = t >> 4;
        const int pr = p / ow;
        const int pc = p % ow;

        float sc[FILT], sh[FILT];
        #pragma unroll
        for (int c = 0; c < FILT; ++c) {
            sc[c] = use_bn ? ss_in[c]        : 1.f;
            sh[c] = use_bn ? ss_in[FILT + c] : 0.f;
        }

        const float* wp = w + (size_t)p * (size_t)(KSZ * KSZ * cin) * FILT;
        __builtin_prefetch(wp, 0, 0);   // global_prefetch_b8 into the weight stream

        float acc[FILT] = {0.f, 0.f, 0.f, 0.f, 0.f, 0.f};
        int k = 0;
        for (int di = 0; di < KSZ; ++di) {
            for (int dj = 0; dj < KSZ; ++dj) {
                const float* irow = in + (((size_t)b * h + (pr + di)) * w_in + (pc + dj)) * cin;
                for (int c = 0; c < cin; ++c, ++k) {
                    float v = irow[c];
                    if (use_bn) {
                        v = fmaf(v, sc[c], sh[c]);
                        v = v > 0.f ? v : 0.f;
                    }
                    const float* wk = wp + k * FILT;
                    #pragma unroll
                    for (int f = 0; f < FILT; ++f) acc[f] = fmaf(v, wk[f], acc[f]);
                }
            }
        }

        float* orow = out + (((size_t)b * oh + pr) * ow + pc) * FILT;
        const float* brow = bias + (size_t)p * FILT;
        #pragma unroll
        for (int f = 0; f < FILT; ++f) {
            float y = acc[f] + brow[f];
            orow[f] = y;
            atomicAdd(&s_sum[f], y);        // ds_add_f32
            atomicAdd(&s_ssq[f], y * y);
        }
    }
    __syncthreads();
    if (threadIdx.x < FILT) {
        atomicAdd(&stats[threadIdx.x],        s_sum[threadIdx.x]);
        atomicAdd(&stats[FILT + threadIdx.x], s_ssq[threadIdx.x]);
    }
}

// ---------------------------------------------------------------------------
// BN stats -> per-channel scale/shift:  scale = gamma*rsqrt(var+eps),
// shift = beta - mean*scale.
// ---------------------------------------------------------------------------
__global__ void bn_finalize(const float* __restrict__ stats,
                            const float* __restrict__ gamma,
                            const float* __restrict__ beta,
                            float* __restrict__ ss_out, float invN)
{
    int c = threadIdx.x;
    if (c >= FILT) return;
    float mean = stats[c] * invN;
    float var  = stats[FILT + c] * invN - mean * mean;
    float s    = gamma[c] * rsqrtf(var + BN_EPS);
    ss_out[c]        = s;
    ss_out[FILT + c] = beta[c] - mean * s;
}

// ---------------------------------------------------------------------------
// Fused Dense(6->256)+ReLU -> Dense(256->128)+ReLU (WMMA fp32) -> avg pool.
// Block = 64 threads = 2 waves; each wave owns a 16-row pixel tile.
// h1 tile (16x256) staged in LDS with 260-float row pitch so the WMMA
// A-fragment ds_load_b64's are bank-conflict-free.
// WMMA f32 16x16x4 layout:
//   A lane<16:  a = {h1[m=lane][4k+0], h1[m=lane][4k+1]},  lane>=16: K+2
//   B lane<16:  b = {B[4k+0][n], B[4k+1][n]},              lane>=16: K+2
//   C vgpr r:   lanes<16 -> (m=r, n), lanes>=16 -> (m=r+8, n)
// ---------------------------------------------------------------------------
#define H1PITCH 260

__global__ void __launch_bounds__(64) dense_pool_kernel(
    const float* __restrict__ y4,     // [B,86,86,6] raw pre-BN layer-5 output
    const float* __restrict__ ss4,    // layer-5 scale[6], shift[6]
    const float* __restrict__ d1w,    // [6,256]
    const float* __restrict__ d1b,    // [256]
    const float* __restrict__ d2w,    // [256,128]
    const float* __restrict__ d2b,    // [128]
    float* __restrict__ pooled)       // [16,128], pre-zeroed
{
    __shared__ float h1s[2][16][H1PITCH];
    __shared__ float x6s[32][8];

    const int tid  = threadIdx.x;
    const int wv   = tid >> 5;
    const int lane = tid & 31;
    const long blockRow0 = (long)blockIdx.x * 32;

    // Stage BN+ReLU'd layer-5 features (6 per row) for this block's 32 rows.
    if (tid < 32) {
        long grow = blockRow0 + tid;
        #pragma unroll
        for (int k = 0; k < FILT; ++k) {
            float v = y4[grow * FILT + k];
            v = fmaf(v, ss4[k], ss4[FILT + k]);
            x6s[tid][k] = v > 0.f ? v : 0.f;
        }
    }
    __syncthreads();

    // Dense1 (K=6): fill wave-private h1 tile in LDS.
    #pragma unroll 1
    for (int j = 0; j < 8; ++j) {
        int col = lane + 32 * j;
        float w6[FILT];
        #pragma unroll
        for (int k = 0; k < FILT; ++k) w6[k] = d1w[k * 256 + col];
        float b1 = d1b[col];
        #pragma unroll 4
        for (int m = 0; m < 16; ++m) {
            float acc = b1;
            #pragma unroll
            for (int k = 0; k < FILT; ++k) acc = fmaf(x6s[wv * 16 + m][k], w6[k], acc);
            h1s[wv][m][col] = acc > 0.f ? acc : 0.f;
        }
    }
    __syncthreads();

    // Dense2 via v_wmma_f32_16x16x4_f32: M=16, N=128 (8 tiles), K=256 (64 steps)
    const int  mlo   = lane & 15;
    const int  khalf = (lane >= 16) ? 2 : 0;
    const int  mhi   = (lane >= 16) ? 8 : 0;

    #pragma unroll 1
    for (int nt = 0; nt < 8; ++nt) {
        const int n = nt * 16 + mlo;
        const float* bcol = d2w + n;
        v8f c = {};
        #pragma unroll 4
        for (int kk = 0; kk < 64; ++kk) {
            int k0 = kk * 4 + khalf;
            v2f a = *(const v2f*)&h1s[wv][mlo][k0];              // ds_load_b64
            v2f b;
            b.x = bcol[(size_t)k0 * 128];
            b.y = bcol[(size_t)(k0 + 1) * 128];
            c = __builtin_amdgcn_wmma_f32_16x16x4_f32(
                    false, a, false, b, (short)0, c, false, false);
        }
        float b2 = d2b[n];
        #pragma unroll
        for (int r = 0; r < 8; ++r) {
            long grow = blockRow0 + wv * 16 + (r + mhi);
            int  bimg = (int)(grow / NPIX4);
            float h2 = c[r] + b2;
            h2 = h2 > 0.f ? h2 : 0.f;
            atomicAdd(&pooled[bimg * 128 + n], h2 * (1.0f / (float)NPIX4));
        }
    }
}

// ---------------------------------------------------------------------------
// Dense(128->10) + softmax; 16 rows, one thread per row.
// ---------------------------------------------------------------------------
__global__ void head_kernel(const float* __restrict__ pooled,
                            const float* __restrict__ d3w,   // [128,10]
                            const float* __restrict__ d3b,   // [10]
                            float* __restrict__ out)          // [16,10]
{
    int b = threadIdx.x;
    if (b >= BATCH) return;
    float logit[10];
    #pragma unroll
    for (int j = 0; j < 10; ++j) {
        float acc = d3b[j];
        for (int k = 0; k < 128; ++k) acc = fmaf(pooled[b * 128 + k], d3w[k * 10 + j], acc);
        logit[j] = acc;
    }
    float mx = logit[0];
    #pragma unroll
    for (int j = 1; j < 10; ++j) mx = logit[j] > mx ? logit[j] : mx;
    float s = 0.f;
    #pragma unroll
    for (int j = 0; j < 10; ++j) { logit[j] = expf(logit[j] - mx); s += logit[j]; }
    float inv = 1.f / s;
    #pragma unroll
    for (int j = 0; j < 10; ++j) out[b * 10 + j] = logit[j] * inv;
}

// ---------------------------------------------------------------------------
// Host-side orchestration (graph-capturable: only async APIs on `stream`).
// ---------------------------------------------------------------------------
static constexpr size_t Y_ELEMS    = (size_t)BATCH * 94 * 94 * FILT;  // 848256
static constexpr size_t OFF_YA     = 0;
static constexpr size_t OFF_YB     = Y_ELEMS * 4;                     // 3,393,024
static constexpr size_t OFF_STATS  = OFF_YB + Y_ELEMS * 4;            // 6,786,048
static constexpr size_t OFF_POOLED = OFF_STATS + 256;
static constexpr size_t OFF_SS     = OFF_POOLED + (size_t)BATCH * 128 * 4;

extern "C" void kernel_launch(void* const* d_in, const int* in_sizes, int n_in,
                              void* d_out, int out_size, void* d_ws, size_t ws_size,
                              hipStream_t stream) {
    (void)in_sizes; (void)n_in; (void)out_size; (void)ws_size;

    const float* x = (const float*)d_in[0];
    const float* lcw[5]; const float* lcb[5]; const float* gam[5]; const float* bet[5];
    for (int i = 0; i < 5; ++i) {
        lcw[i] = (const float*)d_in[1 + i];
        lcb[i] = (const float*)d_in[6 + i];
        gam[i] = (const float*)d_in[11 + i];
        bet[i] = (const float*)d_in[16 + i];
    }
    const float* d1w = (const float*)d_in[21];
    const float* d1b = (const float*)d_in[22];
    const float* d2w = (const float*)d_in[23];
    const float* d2b = (const float*)d_in[24];
    const float* d3w = (const float*)d_in[25];
    const float* d3b = (const float*)d_in[26];

    char*  ws     = (char*)d_ws;
    float* bufs[2] = { (float*)(ws + OFF_YA), (float*)(ws + OFF_YB) };
    float* stats  = (float*)(ws + OFF_STATS);   // 5 layers * 12 floats
    float* pooled = (float*)(ws + OFF_POOLED);  // 16*128
    float* ssbuf  = (float*)(ws + OFF_SS);      // 5 layers * 12 floats

    // Zero BN-stat accumulators and pooled sums (atomically accumulated).
    hipMemsetAsync(ws + OFF_STATS, 0, 256 + (size_t)BATCH * 128 * 4, stream);

    const float* in = x;
    const float* ss_in = ssbuf;   // unused when use_bn == 0
    int use_bn = 0;
    int h = 96, w = 96, cin = 3;

    for (int L = 0; L < 5; ++L) {
        int oh = h - 2, ow = w - 2;
        float* out = bufs[L & 1];
        int nthreads = oh * ow * BATCH;
        int blocks = (nthreads + 255) / 256;
        lc_kernel<<<blocks, 256, 0, stream>>>(in, ss_in, use_bn, lcw[L], lcb[L],
                                              out, stats + L * 12, h, w, cin, oh, ow);
        bn_finalize<<<1, 32, 0, stream>>>(stats + L * 12, gam[L], bet[L],
                                          ssbuf + L * 12,
                                          1.0f / (float)(BATCH * oh * ow));
        in = out; ss_in = ssbuf + L * 12; use_bn = 1;
        h = oh; w = ow; cin = FILT;
    }

    // 118336 rows / 32 rows per block = 3698 blocks (exact)
    dense_pool_kernel<<<(BATCH * NPIX4) / 32, 64, 0, stream>>>(
        in, ss_in, d1w, d1b, d2w, d2b, pooled);

    head_kernel<<<1, 32, 0, stream>>>(pooled, d3w, d3b, (float*)d_out);
}